// ModMMD_71880572666159
// MI455X (gfx1250) — compile-verified
//
#include <hip/hip_runtime.h>
#include <hip/hip_bf16.h>

typedef __attribute__((ext_vector_type(2))) float v2f;
typedef __attribute__((ext_vector_type(4))) float v4f;
typedef __attribute__((ext_vector_type(8))) float v8f;

#define D 128            // feature dim (K)
#define KSTEPS (D / 4)   // 32 wmma_f32_16x16x4 steps per tile
#define JT 8             // j-tiles (of 16 cols) per wave
#define TPB 256          // threads per block (8 waves)

// ---------------------------------------------------------------------------
// Pass 1: row squared-norms. One thread per row (x rows then y rows).
// ---------------------------------------------------------------------------
__global__ void mmd_rownorm_kernel(const float* __restrict__ x,
                                   const float* __restrict__ y,
                                   float* __restrict__ norms, int n, int m) {
    int t = blockIdx.x * blockDim.x + threadIdx.x;
    if (t >= n + m) return;
    const float* src = (t < n) ? (x + (size_t)t * D) : (y + (size_t)(t - n) * D);
    float s = 0.0f;
#pragma unroll
    for (int k = 0; k < D / 4; ++k) {
        v4f v = *(const v4f*)(src + 4 * k);
        s += v.x * v.x + v.y * v.y + v.z * v.z + v.w * v.w;
    }
    norms[t] = s;
}

// ---------------------------------------------------------------------------
// Pass 2: tiled Gram via V_WMMA_F32_16X16X4_F32 + exp + hierarchical reduce.
//
// Wave -> (matrix, i-tile, j-group of JT tiles). The 8 waves of a block share
// one j-group (B strips dedup through WGP$); A strip is per-wave in VGPRs.
// Per j-tile: clause-load the full B strip (32 x float2), one wait, then two
// independent interleaved WMMA accumulation chains (even/odd k-steps).
//
// f32 16x4 A layout (ISA 7.12.2): lane L (L<16) holds A[M=L][k0+{0,1}],
// lane L+16 holds A[M=L][k0+{2,3}]. B 4x16 mirrors with N instead of M.
// C/D layout: VGPR v, lane L: row = v + 8*(L>>4), col = L&15.
// ---------------------------------------------------------------------------
__global__ void __launch_bounds__(TPB)
mmd_wmma_kernel(const float* __restrict__ x, const float* __restrict__ y,
                const float* __restrict__ xn, const float* __restrict__ yn,
                float* __restrict__ partials, int n,
                int wavesPerMat, int itiles,
                float coef_xx, float coef_xy, float coef_yy) {
    const int lane = threadIdx.x & 31;
    const int w    = (blockIdx.x * TPB + threadIdx.x) >> 5;

    const int mat    = w / wavesPerMat;          // 0:xx 1:xy 2:yy (wave-uniform)
    const int rem    = w - mat * wavesPerMat;
    const int jgroup = rem / itiles;             // shared by the block's 8 waves
    const int itile  = rem - jgroup * itiles;    // consecutive per wave

    const float* P;  const float* Q;  const float* Pn;  const float* Qn;  float coef;
    if (mat == 0)      { P = x; Q = x; Pn = xn; Qn = xn; coef = coef_xx; }
    else if (mat == 1) { P = x; Q = y; Pn = xn; Qn = yn; coef = coef_xy; }
    else               { P = y; Q = y; Pn = yn; Qn = yn; coef = coef_yy; }

    const int i0     = itile * 16;
    const int j0base = jgroup * (JT * 16);
    const int r      = lane & 15;        // row/col within tile for A/B loads
    const int koff   = (lane >> 4) * 2;  // k sub-offset {0,2}
    const int hi     = lane >> 4;        // C-half selector

    // Preload A strip: 16 rows x 128 k -> 32 x float2 per lane (64 VGPRs).
    const float* arow = P + (size_t)(i0 + r) * D + koff;
    v2f areg[KSTEPS];
#pragma unroll
    for (int s = 0; s < KSTEPS; ++s) areg[s] = *(const v2f*)(arow + 4 * s);

    // Row norms for the 8 C rows this lane produces.
    float pn[8];
#pragma unroll
    for (int v = 0; v < 8; ++v) pn[v] = Pn[i0 + v + 8 * hi];

    float acc = 0.0f;
    for (int jt = 0; jt < JT; ++jt) {
        const int j0 = j0base + jt * 16;
        const float* brow = Q + (size_t)(j0 + r) * D + koff;

        // Batched B-strip load: one clause of 32 global_load_b64, one wait.
        v2f breg[KSTEPS];
#pragma unroll
        for (int s = 0; s < KSTEPS; ++s) breg[s] = *(const v2f*)(brow + 4 * s);

        // Two independent WMMA chains to keep the matrix pipe busy.
        v8f c0 = {}, c1 = {};
#pragma unroll
        for (int s = 0; s < KSTEPS / 2; ++s) {
            c0 = __builtin_amdgcn_wmma_f32_16x16x4_f32(
                     false, areg[2 * s],     false, breg[2 * s],
                     (short)0, c0, false, false);
            c1 = __builtin_amdgcn_wmma_f32_16x16x4_f32(
                     false, areg[2 * s + 1], false, breg[2 * s + 1],
                     (short)0, c1, false, false);
        }

        const float q2 = Qn[j0 + r];
#pragma unroll
        for (int v = 0; v < 8; ++v) {
            float dot = c0[v] + c1[v];
            float d2  = pn[v] + q2 - 2.0f * dot;
            d2 = fmaxf(d2, 0.0f);          // cdist**2 clamp
            acc += __expf(-d2);            // bandwidth == 1.0 for all three
        }
    }

    // Wave32 reduction, coefficient applied per wave.
#pragma unroll
    for (int off = 16; off >= 1; off >>= 1) acc += __shfl_xor(acc, off, 32);

    __shared__ float sdata[TPB / 32];
    if (lane == 0) sdata[threadIdx.x >> 5] = acc * coef;
    __syncthreads();
    if (threadIdx.x == 0) {
        float s = 0.0f;
#pragma unroll
        for (int i = 0; i < TPB / 32; ++i) s += sdata[i];
        partials[blockIdx.x] = s;
    }
}

// ---------------------------------------------------------------------------
// Pass 3: deterministic fixed-order reduction of block partials -> scalar.
// ---------------------------------------------------------------------------
__global__ void mmd_reduce_kernel(const float* __restrict__ partials, int n,
                                  float* __restrict__ out) {
    __shared__ float sm[TPB];
    float s = 0.0f;
    for (int i = threadIdx.x; i < n; i += TPB) s += partials[i];
    sm[threadIdx.x] = s;
    __syncthreads();
    for (int off = TPB / 2; off >= 1; off >>= 1) {
        if ((int)threadIdx.x < off) sm[threadIdx.x] += sm[threadIdx.x + off];
        __syncthreads();
    }
    if (threadIdx.x == 0) out[0] = sm[0];
}

extern "C" void kernel_launch(void* const* d_in, const int* in_sizes, int n_in,
                              void* d_out, int out_size, void* d_ws, size_t ws_size,
                              hipStream_t stream) {
    const float* x = (const float*)d_in[0];
    const float* y = (const float*)d_in[1];
    float* out = (float*)d_out;

    const int n = in_sizes[0] / D;   // 8192
    const int m = in_sizes[1] / D;   // 8192 (n == m assumed, as in reference)

    // Workspace layout: [x2 (n)] [y2 (m)] [block partials]
    float* xn = (float*)d_ws;
    float* yn = xn + n;
    float* partials = yn + m;

    const int itiles  = n / 16;               // 512
    const int jgroups = m / (16 * JT);        // 64
    const int wavesPerMat = itiles * jgroups; // 32768
    const long long totalThreads = 3LL * wavesPerMat * 32;
    const int nblocks = (int)(totalThreads / TPB);  // 12288

    const double dn = (double)n, dm = (double)m;
    const float coef_xx = (float)(0.5 / (dn * (dn - 1.0)));   // AXX * xx_scale
    const float coef_xy = (float)(1.0 * (-2.0 / (dn * dm)));  // AXY * xy_scale
    const float coef_yy = (float)(0.5 / (dm * (dm - 1.0)));   // AYY * yy_scale

    mmd_rownorm_kernel<<<(n + m + TPB - 1) / TPB, TPB, 0, stream>>>(x, y, xn, n, m);
    mmd_wmma_kernel<<<nblocks, TPB, 0, stream>>>(x, y, xn, yn, partials, n,
                                                 wavesPerMat, itiles,
                                                 coef_xx, coef_xy, coef_yy);
    mmd_reduce_kernel<<<1, TPB, 0, stream>>>(partials, nblocks, out);
}